// Warp_33062658244995
// MI455X (gfx1250) — compile-verified
//
#include <hip/hip_runtime.h>

// Bilinear flow-warp, NHWC, B=4 H=512 W=512 C=64, f32.
// Memory-bound: map 16 lanes -> 1 pixel, 1 float4 (4 channels) per lane.
// Every neighbor fetch from a half-wave is a contiguous 256B global_load_b128.

typedef float f32x4 __attribute__((ext_vector_type(4)));
typedef float f32x2 __attribute__((ext_vector_type(2)));

namespace {
constexpr int H = 512;
constexpr int W = 512;
constexpr int C = 64;

__device__ __forceinline__ float clampf(float v, float lo, float hi) {
    return fminf(fmaxf(v, lo), hi);
}
}  // namespace

__global__ __launch_bounds__(256) void warp_bilinear_kernel(
    const float* __restrict__ xin,
    const float* __restrict__ flow,
    float* __restrict__ out,
    int npix)
{
    const int tid = blockIdx.x * 256 + threadIdx.x;
    const int pix = tid >> 4;            // 16 threads per pixel
    if (pix >= npix) return;
    const int cv = (tid & 15) << 2;      // channel start: 0,4,...,60

    // Pull the flow stream ahead into L2 (gfx1250 global_prefetch_b8).
    // Flow reads gate the data-dependent gathers, so lead them.
    const int pf = pix + 16384;
    if (pf < npix) __builtin_prefetch(flow + 2 * (size_t)pf, 0, 1);

    const f32x2 fl = ((const f32x2*)flow)[pix];

    // H = W = 512 -> pure shifts/masks.
    const int wx = pix & (W - 1);
    const int hy = (pix >> 9) & (H - 1);
    const int b  = pix >> 18;

    const float qx = (float)wx + fl.x;
    const float qy = (float)hy + fl.y;

    // tfa semantics: clamp floor to [0, size-2] FIRST, then alpha vs clamped floor.
    const float fxf = clampf(floorf(qx), 0.0f, (float)(W - 2));
    const float fyf = clampf(floorf(qy), 0.0f, (float)(H - 2));
    const float ax  = clampf(qx - fxf, 0.0f, 1.0f);
    const float ay  = clampf(qy - fyf, 0.0f, 1.0f);
    const int fx = (int)fxf;
    const int fy = (int)fyf;

    const float* base = xin + ((((size_t)b * H + fy) * W + fx) * C) + cv;
    const f32x4 tl = *(const f32x4*)(base);
    const f32x4 tr = *(const f32x4*)(base + C);
    const f32x4 bl = *(const f32x4*)(base + (size_t)W * C);
    const f32x4 br = *(const f32x4*)(base + (size_t)W * C + C);

    f32x4 o;
    {
        float top, bot;
        top = fmaf(ax, tr.x - tl.x, tl.x);
        bot = fmaf(ax, br.x - bl.x, bl.x);
        o.x = fmaf(ay, bot - top, top);

        top = fmaf(ax, tr.y - tl.y, tl.y);
        bot = fmaf(ax, br.y - bl.y, bl.y);
        o.y = fmaf(ay, bot - top, top);

        top = fmaf(ax, tr.z - tl.z, tl.z);
        bot = fmaf(ax, br.z - bl.z, bl.z);
        o.z = fmaf(ay, bot - top, top);

        top = fmaf(ax, tr.w - tl.w, tl.w);
        bot = fmaf(ax, br.w - bl.w, bl.w);
        o.w = fmaf(ay, bot - top, top);
    }

    // Output is write-once: non-temporal b128 store keeps the 192MB L2 for
    // the gather working set.
    __builtin_nontemporal_store(o, (f32x4*)(out + (size_t)pix * C + cv));
}

extern "C" void kernel_launch(void* const* d_in, const int* in_sizes, int n_in,
                              void* d_out, int out_size, void* d_ws, size_t ws_size,
                              hipStream_t stream)
{
    const float* x    = (const float*)d_in[0];
    const float* flow = (const float*)d_in[1];
    float* out        = (float*)d_out;

    const int npix     = in_sizes[1] / 2;   // B*H*W = 1,048,576
    const int nthreads = npix * 16;         // 16 lanes (64 ch / float4) per pixel
    const int blocks   = (nthreads + 255) / 256;

    warp_bilinear_kernel<<<blocks, 256, 0, stream>>>(x, flow, out, npix);
}